// PostProcessor_18408229830944
// MI455X (gfx1250) — compile-verified
//
#include <hip/hip_runtime.h>
#include <stdint.h>

// ---------------------------------------------------------------------------
// Stereo detection post-processor for MI455X (gfx1250, wave32).
// Pipeline: decode -> per-class greedy stereo-NMS -> gather kept feats ->
//           top-100 across classes -> WMMA one-hot gather into [100,17] out.
// CDNA5 paths used:
//   * global_load_async_to_lds_b128 + s_wait_asynccnt (proposal staging)
//   * v_wmma_f32_16x16x4_f32 (exact f32 selection-matrix gather)
// ---------------------------------------------------------------------------

#define NEGV      (-1.0e30f)
#define NEGHALF   (-5.0e29f)
#define NEGINF    (-__builtin_inff())
#define DW_CLAMP  4.135166556742356f      // log(1000/16)
#define IMG_W1    1279.0f
#define IMG_H1    383.0f
#define PI_F      3.14159265358979323846f
#define BIN_SIZE  (2.0f * PI_F / 10.0f)

typedef float v2f __attribute__((ext_vector_type(2)));
typedef float v8f __attribute__((ext_vector_type(8)));

// ---------------------------------------------------------------------------
// Kernel 1: decode. Softmax scores (thresholded), decoded L/R boxes per class.
// Proposals staged block-wide into LDS via async-to-LDS loads.
// ---------------------------------------------------------------------------
__global__ void decode_kernel(const float* __restrict__ class_logits,
                              const float* __restrict__ bbL,
                              const float* __restrict__ bbR,
                              const float* __restrict__ propL,
                              const float* __restrict__ propR,
                              float* __restrict__ clsScores,
                              float4* __restrict__ boxesL,
                              float4* __restrict__ boxesR,
                              int N) {
  __shared__ float4 spl[256];
  __shared__ float4 spr[256];
  const int tid = threadIdx.x;
  const int i = blockIdx.x * 256 + tid;

  // Async staging: LDS dest address (wave-relative low 32 bits of flat addr),
  // per-lane 32-bit byte offset against a uniform SGPR base (GVS mode).
  unsigned la = (unsigned)(uintptr_t)&spl[tid];
  unsigned lb = (unsigned)(uintptr_t)&spr[tid];
  unsigned voff = (unsigned)i * 16u;
  if (i < N) {
    asm volatile("global_load_async_to_lds_b128 %0, %1, %2"
                 :: "v"(la), "v"(voff), "s"(propL) : "memory");
    asm volatile("global_load_async_to_lds_b128 %0, %1, %2"
                 :: "v"(lb), "v"(voff), "s"(propR) : "memory");
  }
  asm volatile("s_wait_asynccnt 0" ::: "memory");
  __syncthreads();
  if (i >= N) return;

  // softmax over 4 class logits
  const float* lg = class_logits + (size_t)i * 4;
  float a0 = lg[0], a1 = lg[1], a2 = lg[2], a3 = lg[3];
  float m = fmaxf(fmaxf(a0, a1), fmaxf(a2, a3));
  float e0 = expf(a0 - m), e1 = expf(a1 - m), e2 = expf(a2 - m), e3 = expf(a3 - m);
  float inv = 1.0f / (e0 + e1 + e2 + e3);
  float s1 = e1 * inv, s2 = e2 * inv, s3 = e3 * inv;
  clsScores[0 * (size_t)N + i] = (s1 > 0.05f) ? s1 : NEGV;
  clsScores[1 * (size_t)N + i] = (s2 > 0.05f) ? s2 : NEGV;
  clsScores[2 * (size_t)N + i] = (s3 > 0.05f) ? s3 : NEGV;

  float4 pl = spl[tid];
  float4 pr = spr[tid];
  float wL = pl.z - pl.x + 1.0f, hL = pl.w - pl.y + 1.0f;
  float cxL = pl.x + 0.5f * wL, cyL = pl.y + 0.5f * hL;
  float wR = pr.z - pr.x + 1.0f, hR = pr.w - pr.y + 1.0f;
  float cxR = pr.x + 0.5f * wR, cyR = pr.y + 0.5f * hR;

  const float* rowL = bbL + (size_t)i * 16;
  const float* rowR = bbR + (size_t)i * 16;
#pragma unroll
  for (int c = 1; c < 4; ++c) {
    // left
    {
      float dx = rowL[4 * c + 0] / 10.0f;
      float dy = rowL[4 * c + 1] / 10.0f;
      float dw = fminf(rowL[4 * c + 2] / 5.0f, DW_CLAMP);
      float dh = fminf(rowL[4 * c + 3] / 5.0f, DW_CLAMP);
      float pcx = dx * wL + cxL, pcy = dy * hL + cyL;
      float pw = expf(dw) * wL, ph = expf(dh) * hL;
      float x1 = fminf(fmaxf(pcx - 0.5f * pw, 0.0f), IMG_W1);
      float y1 = fminf(fmaxf(pcy - 0.5f * ph, 0.0f), IMG_H1);
      float x2 = fminf(fmaxf(pcx + 0.5f * pw - 1.0f, 0.0f), IMG_W1);
      float y2 = fminf(fmaxf(pcy + 0.5f * ph - 1.0f, 0.0f), IMG_H1);
      boxesL[(size_t)(c - 1) * N + i] = make_float4(x1, y1, x2, y2);
    }
    // right
    {
      float dx = rowR[4 * c + 0] / 10.0f;
      float dy = rowR[4 * c + 1] / 10.0f;
      float dw = fminf(rowR[4 * c + 2] / 5.0f, DW_CLAMP);
      float dh = fminf(rowR[4 * c + 3] / 5.0f, DW_CLAMP);
      float pcx = dx * wR + cxR, pcy = dy * hR + cyR;
      float pw = expf(dw) * wR, ph = expf(dh) * hR;
      float x1 = fminf(fmaxf(pcx - 0.5f * pw, 0.0f), IMG_W1);
      float y1 = fminf(fmaxf(pcy - 0.5f * ph, 0.0f), IMG_H1);
      float x2 = fminf(fmaxf(pcx + 0.5f * pw - 1.0f, 0.0f), IMG_W1);
      float y2 = fminf(fmaxf(pcy + 0.5f * ph - 1.0f, 0.0f), IMG_H1);
      boxesR[(size_t)(c - 1) * N + i] = make_float4(x1, y1, x2, y2);
    }
  }
}

// ---------------------------------------------------------------------------
// Kernel 2: greedy stereo NMS. One 1024-thread (32-wave) block per class.
// Serial 100 iterations of (argmax, suppress by max(iouL,iouR) > 0.5).
// First-occurrence argmax tie-break matches jnp.argmax.
// ---------------------------------------------------------------------------
#define NMS_T 1024
__global__ void nms_kernel(float* __restrict__ clsScores,
                           const float4* __restrict__ boxesL,
                           const float4* __restrict__ boxesR,
                           int* __restrict__ keep, int N) {
  __shared__ float sv[NMS_T];
  __shared__ int si[NMS_T];
  __shared__ float4 sPL, sPR;
  __shared__ int sIdx, sOk;
  const int c = blockIdx.x;
  const int tid = threadIdx.x;
  float* s = clsScores + (size_t)c * N;
  const float4* bl = boxesL + (size_t)c * N;
  const float4* br = boxesR + (size_t)c * N;

  for (int iter = 0; iter < 100; ++iter) {
    // per-thread scan over strided chunk (indices increasing => first max kept)
    float bv = NEGINF;
    int bi = 0x7fffffff;
    for (int i = tid; i < N; i += NMS_T) {
      float v = s[i];
      if (v > bv) { bv = v; bi = i; }
    }
    sv[tid] = bv; si[tid] = bi;
    __syncthreads();
    for (int o = NMS_T / 2; o > 0; o >>= 1) {
      if (tid < o) {
        float v2 = sv[tid + o]; int i2 = si[tid + o];
        if (v2 > sv[tid] || (v2 == sv[tid] && i2 < si[tid])) { sv[tid] = v2; si[tid] = i2; }
      }
      __syncthreads();
    }
    if (tid == 0) {
      int idx = si[0];
      float v = sv[0];
      int ok = (v > NEGHALF) ? 1 : 0;
      keep[c * 100 + iter] = ok ? idx : -1;
      sIdx = idx; sOk = ok;
      sPL = bl[idx]; sPR = br[idx];
    }
    __syncthreads();
    const int idx = sIdx;
    const int ok = sOk;
    const float4 PL = sPL, PR = sPR;
    if (ok) {
      float aL = (PL.z - PL.x + 1.0f) * (PL.w - PL.y + 1.0f);
      float aR = (PR.z - PR.x + 1.0f) * (PR.w - PR.y + 1.0f);
      for (int i = tid; i < N; i += NMS_T) {
        float4 b = bl[i];
        float ar = (b.z - b.x + 1.0f) * (b.w - b.y + 1.0f);
        float ix1 = fmaxf(b.x, PL.x), iy1 = fmaxf(b.y, PL.y);
        float ix2 = fminf(b.z, PL.z), iy2 = fminf(b.w, PL.w);
        float inter = fmaxf(ix2 - ix1 + 1.0f, 0.0f) * fmaxf(iy2 - iy1 + 1.0f, 0.0f);
        float iouL = inter / (ar + aL - inter);
        b = br[i];
        ar = (b.z - b.x + 1.0f) * (b.w - b.y + 1.0f);
        ix1 = fmaxf(b.x, PR.x); iy1 = fmaxf(b.y, PR.y);
        ix2 = fminf(b.z, PR.z); iy2 = fminf(b.w, PR.w);
        inter = fmaxf(ix2 - ix1 + 1.0f, 0.0f) * fmaxf(iy2 - iy1 + 1.0f, 0.0f);
        float iouR = inter / (ar + aR - inter);
        if (fmaxf(iouL, iouR) > 0.5f) s[i] = NEGV;
      }
    }
    __threadfence_block();
    __syncthreads();
    if (tid == 0) s[idx] = NEGV;
    __threadfence_block();
    __syncthreads();
  }
}

// ---------------------------------------------------------------------------
// Kernel 3: gather 300 kept candidates -> Feat[300,16] and ksArr[300].
// Centers/dims/rot are recomputed on the fly for just the kept indices.
// ---------------------------------------------------------------------------
__global__ void gather_kernel(const float* __restrict__ class_logits,
                              const float* __restrict__ centerL,
                              const float* __restrict__ centerR,
                              const float* __restrict__ hwl,
                              const float* __restrict__ alpha_logit,
                              const float* __restrict__ alpha_reg,
                              const float4* __restrict__ propL,
                              const float4* __restrict__ propR,
                              const float4* __restrict__ boxesL,
                              const float4* __restrict__ boxesR,
                              const int* __restrict__ keep,
                              float* __restrict__ Feat,
                              float* __restrict__ ksArr, int N) {
  const int c1 = blockIdx.x;       // 0..2, class = c1+1
  const int j = threadIdx.x;
  if (j >= 100) return;
  const int r = c1 * 100 + j;
  const int k = keep[r];
  const int safe = (k > 0) ? k : 0;
  const int cls = c1 + 1;

  // score (recompute softmax; NMS mutated the stored copy)
  const float* lg = class_logits + (size_t)safe * 4;
  float a0 = lg[0], a1 = lg[1], a2 = lg[2], a3 = lg[3];
  float m = fmaxf(fmaxf(a0, a1), fmaxf(a2, a3));
  float e[4] = {expf(a0 - m), expf(a1 - m), expf(a2 - m), expf(a3 - m)};
  float inv = 1.0f / (e[0] + e[1] + e[2] + e[3]);
  float sc = e[cls] * inv;
  float th = (sc > 0.05f) ? sc : NEGV;
  ksArr[r] = (k >= 0) ? th : NEGV;

  float4 bl = boxesL[(size_t)c1 * N + safe];
  float4 br = boxesR[(size_t)c1 * N + safe];

  // centers
  float4 pL = propL[safe];
  float wL = pL.z - pL.x + 1.0f, hL = pL.w - pL.y + 1.0f;
  float cxL = pL.x + 0.5f * wL, cyL = pL.y + 0.5f * hL;
  float pcxL = centerL[(size_t)safe * 8 + 2 * cls] / 10.0f * wL + cxL;
  float pcyL = centerL[(size_t)safe * 8 + 2 * cls + 1] / 10.0f * hL + cyL;
  float4 pR = propR[safe];
  float wR = pR.z - pR.x + 1.0f, hR = pR.w - pR.y + 1.0f;
  float cxR = pR.x + 0.5f * wR, cyR = pR.y + 0.5f * hR;
  float pcxR = centerR[(size_t)safe * 8 + 2 * cls] / 10.0f * wR + cxR;
  float pcyR = centerR[(size_t)safe * 8 + 2 * cls + 1] / 10.0f * hR + cyR;

  // dims (KITTI mean h,w,l)
  const float MEAN[3] = {1.53f, 1.63f, 3.88f};
  float d0 = expf(hwl[(size_t)safe * 12 + 3 * cls + 0]) * MEAN[0];
  float d1 = expf(hwl[(size_t)safe * 12 + 3 * cls + 1]) * MEAN[1];
  float d2 = expf(hwl[(size_t)safe * 12 + 3 * cls + 2]) * MEAN[2];

  // rotation: argmax over 10 alpha bins (first max), then bin residual
  const float* al = alpha_logit + (size_t)safe * 10;
  int lab = 0; float bv = al[0];
#pragma unroll
  for (int t = 1; t < 10; ++t) {
    float v = al[t];
    if (v > bv) { bv = v; lab = t; }
  }
  float res = alpha_reg[(size_t)safe * 40 + 10 * cls + lab];
  float rot = ((float)lab + res) * BIN_SIZE - PI_F;

  float* f = Feat + (size_t)r * 16;
  f[0] = bl.x; f[1] = bl.y; f[2] = bl.z; f[3] = bl.w;
  f[4] = br.x; f[5] = br.y; f[6] = br.z; f[7] = br.w;
  f[8] = pcxL; f[9] = pcyL; f[10] = pcxR; f[11] = pcyR;
  f[12] = d0; f[13] = d1; f[14] = d2; f[15] = rot;
}

// ---------------------------------------------------------------------------
// Kernel 4: top-100 of 300 scores (descending, stable ties by index == 100
// repeated first-occurrence argmaxes). Also zero-pads Feat rows 300..303 and
// writes the score column (out[:,16]).
// ---------------------------------------------------------------------------
__global__ void topk_kernel(const float* __restrict__ ksArr,
                            float* __restrict__ Feat,
                            int* __restrict__ topIdx,
                            int* __restrict__ topValid,
                            float* __restrict__ out) {
  __shared__ float sv[512];
  __shared__ int si[512];
  const int tid = threadIdx.x;

  if (tid >= 300 && tid < 304) {
    float* f = Feat + (size_t)tid * 16;
#pragma unroll
    for (int t = 0; t < 16; ++t) f[t] = 0.0f;
  }
  float myv = (tid < 300) ? ksArr[tid] : NEGINF;

  for (int iter = 0; iter < 100; ++iter) {
    sv[tid] = myv; si[tid] = tid;
    __syncthreads();
    for (int o = 256; o > 0; o >>= 1) {
      if (tid < o) {
        float v2 = sv[tid + o]; int i2 = si[tid + o];
        if (v2 > sv[tid] || (v2 == sv[tid] && i2 < si[tid])) { sv[tid] = v2; si[tid] = i2; }
      }
      __syncthreads();
    }
    int idx = si[0];
    float best = sv[0];
    __syncthreads();
    if (tid == 0) {
      int val = (best > NEGHALF) ? 1 : 0;
      topIdx[iter] = idx;
      topValid[iter] = val;
      out[iter * 17 + 16] = val ? best : 0.0f;
    }
    if (tid == idx) myv = NEGINF;
    __syncthreads();
  }
}

// ---------------------------------------------------------------------------
// Kernel 5: WMMA one-hot gather. Out[100,16] = S[100,304] @ Feat[304,16]
// with S one-hot (x validity mask) -> exact in f32 via V_WMMA_F32_16X16X4_F32.
// 7 M-tiles (one wave each), K = 304 in 76 steps of 4.
// ---------------------------------------------------------------------------
__global__ void wmma_gather_kernel(const float* __restrict__ Feat,
                                   const int* __restrict__ topIdx,
                                   const int* __restrict__ topValid,
                                   float* __restrict__ out) {
  const int lane = threadIdx.x;          // 32 lanes = one wave
  const int m0 = blockIdx.x * 16;        // 7 blocks cover 112 >= 100 rows
  const int rsel = lane & 15;
  const int rowA = m0 + rsel;            // A layout: lanes 0-15 K={0,1}, 16-31 K={2,3}
  int tI = -1;
  if (rowA < 100) {
    if (topValid[rowA]) tI = topIdx[rowA];
  }
  const int hi = (lane >= 16) ? 2 : 0;
  const int colB = lane & 15;            // B layout: N = lane%16, K rows split by half-wave

  v8f acc = {};
  for (int kk = 0; kk < 76; ++kk) {
    int k0 = kk * 4 + hi;
    v2f a, b;
    a.x = (tI == k0) ? 1.0f : 0.0f;
    a.y = (tI == k0 + 1) ? 1.0f : 0.0f;
    b.x = Feat[(size_t)k0 * 16 + colB];
    b.y = Feat[(size_t)(k0 + 1) * 16 + colB];
    acc = __builtin_amdgcn_wmma_f32_16x16x4_f32(
        /*neg_a=*/false, a, /*neg_b=*/false, b,
        /*c_mod=*/(short)0, acc, /*reuse_a=*/false, /*reuse_b=*/false);
  }

  // D layout: VGPR r -> M = r (lanes 0-15) or r+8 (lanes 16-31), N = lane%16
  const int mbase = m0 + ((lane >= 16) ? 8 : 0);
#pragma unroll
  for (int r = 0; r < 8; ++r) {
    int M = mbase + r;
    if (M < 100) out[(size_t)M * 17 + colB] = acc[r];
  }
}

// ---------------------------------------------------------------------------
// Launch
// ---------------------------------------------------------------------------
extern "C" void kernel_launch(void* const* d_in, const int* in_sizes, int n_in,
                              void* d_out, int out_size, void* d_ws, size_t ws_size,
                              hipStream_t stream) {
  (void)n_in; (void)out_size; (void)ws_size;
  const int N = in_sizes[0] / 4;  // class_logits is [N,4]

  const float* class_logits = (const float*)d_in[0];
  const float* bbox_l       = (const float*)d_in[1];
  const float* bbox_r       = (const float*)d_in[2];
  const float* center_l     = (const float*)d_in[3];
  const float* center_r     = (const float*)d_in[4];
  const float* hwl          = (const float*)d_in[5];
  const float* alpha_logit  = (const float*)d_in[6];
  const float* alpha_reg    = (const float*)d_in[7];
  const float* prop_l       = (const float*)d_in[8];
  const float* prop_r       = (const float*)d_in[9];
  float* out = (float*)d_out;

  // workspace carve-up (~27N floats + ~30 KB; N=131072 -> ~14.3 MB)
  float* ws = (float*)d_ws;
  float*  clsScores = ws;                               // 3N floats (mutable)
  float4* boxesL    = (float4*)(ws + 3 * (size_t)N);    // 3N float4
  float4* boxesR    = (float4*)(ws + 15 * (size_t)N);   // 3N float4
  int*    keep      = (int*)(ws + 27 * (size_t)N);      // 300 (pad 304)
  int*    topIdx    = keep + 304;                       // 100 (pad 112)
  int*    topValid  = topIdx + 112;                     // 100 (pad 112)
  float*  Feat      = (float*)(topValid + 112);         // 304*16
  float*  ksArr     = Feat + 304 * 16;                  // 300 (pad 304)

  decode_kernel<<<(N + 255) / 256, 256, 0, stream>>>(
      class_logits, bbox_l, bbox_r, prop_l, prop_r,
      clsScores, boxesL, boxesR, N);

  nms_kernel<<<3, NMS_T, 0, stream>>>(clsScores, boxesL, boxesR, keep, N);

  gather_kernel<<<3, 128, 0, stream>>>(
      class_logits, center_l, center_r, hwl, alpha_logit, alpha_reg,
      (const float4*)prop_l, (const float4*)prop_r,
      boxesL, boxesR, keep, Feat, ksArr, N);

  topk_kernel<<<1, 512, 0, stream>>>(ksArr, Feat, topIdx, topValid, out);

  wmma_gather_kernel<<<7, 32, 0, stream>>>(Feat, topIdx, topValid, out);
}